// Dynamics_53884659696325
// MI455X (gfx1250) — compile-verified
//
#include <hip/hip_runtime.h>

// ----- CDNA5 WMMA vector types (wave32) -----
typedef __attribute__((ext_vector_type(2))) float v2f;  // 16x4 (A) / 4x16 (B) f32 operand: 2 VGPRs
typedef __attribute__((ext_vector_type(8))) float v8f;  // 16x16 f32 C/D: 8 VGPRs

// ----- Physics constants (match reference, evaluated in double then narrowed) -----
#define BOOST_ACCEL   0.18f
#define FRAME_TIME    0.1f
#define GRAVITY_ACCEL 0.12f
// C_D = pi/16 * (0.05*15) * 0.05 = 0.0073631077818510775...
#define C_D           0.0073631077818510775f

// One wave32 does everything:
//  1) VALU: ns = state + gravity + BOOST*dt*(state_tensor @ action)   (sin/cos -> VALU/libm)
//  2) WMMA (f32, 16x16x4, exact): out = (dt*E) @ ns + ns  == step_mat @ ns
//     since step_mat = I + dt*E with E[0][2]=E[1][3]=1 (only K-columns 2,3 -> fits K=4).
__global__ __launch_bounds__(32) void Dynamics_53884659696325_kernel(
    const float* __restrict__ state,
    const float* __restrict__ action,
    float* __restrict__ out) {
  const int lane = threadIdx.x;  // 0..31, single wave32; EXEC all ones

  // Broadcast loads: every lane reads the 7 scalars (hits L0/L2).
  const float st0 = state[0];
  const float st1 = state[1];
  const float st2 = state[2];
  const float st3 = state[3];
  const float st4 = state[4];
  const float a0  = action[0];
  const float a1  = action[1];

  // Shared range reduction for sin+cos (libm accuracy, matches jnp.sin/cos closely).
  float s, c;
  sincosf(st4, &s, &c);

  const float k = BOOST_ACCEL * FRAME_TIME;
  // ns = state + delta_state + delta_state_gravity
  const float ns0 = st0 + k * (-0.5f * FRAME_TIME * s) * a0;
  const float ns1 = st1 + k * ( 0.5f * FRAME_TIME * c) * a0 - GRAVITY_ACCEL * FRAME_TIME;
  const float ns2 = st2 + k * (-s - C_D * st2 * st2) * a0;
  const float ns3 = st3 + k * ( c - C_D * st3 * st3) * a0;
  const float ns4 = st4 + k * (1.0f / FRAME_TIME) * a1;

  // ---- A matrix (16x4 f32): lanes 0-15 hold K=0 (v0) / K=1 (v1) for row M=lane;
  //      lanes 16-31 hold K=2 (v0) / K=3 (v1) for row M=lane-16.
  // Nonzeros: A[0][2] = A[1][3] = FRAME_TIME.
  v2f A;
  A.x = (lane == 16) ? FRAME_TIME : 0.0f;  // A[M=0][K=2]
  A.y = (lane == 17) ? FRAME_TIME : 0.0f;  // A[M=1][K=3]

  // ---- B matrix (4x16 f32): v0 lanes 0-15 = row K=0 (N=lane), lanes 16-31 = row K=2;
  //      v1 lanes 0-15 = row K=1, lanes 16-31 = row K=3. Only column N=0 is used.
  v2f B;
  B.x = (lane == 0) ? ns0 : ((lane == 16) ? ns2 : 0.0f);  // K=0 / K=2, N=0
  B.y = (lane == 0) ? ns1 : ((lane == 16) ? ns3 : 0.0f);  // K=1 / K=3, N=0

  // ---- C matrix (16x16 f32): VGPR r, lanes 0-15 = row M=r (N=lane), lanes 16-31 = row M=r+8.
  // Column N=0 (lane 0), rows 0..4 carry ns.
  v8f C = {};
  C[0] = (lane == 0) ? ns0 : 0.0f;
  C[1] = (lane == 0) ? ns1 : 0.0f;
  C[2] = (lane == 0) ? ns2 : 0.0f;
  C[3] = (lane == 0) ? ns3 : 0.0f;
  C[4] = (lane == 0) ? ns4 : 0.0f;

  // D = A*B + C : exact f32 (RNE).  D[m][0] = step_mat @ ns, rows 0..4.
  // 8-arg form: (neg_a, A, neg_b, B, c_mod, C, reuse_a, reuse_b)
  v8f D = __builtin_amdgcn_wmma_f32_16x16x4_f32(
      /*neg_a=*/false, A, /*neg_b=*/false, B,
      /*c_mod=*/(short)0, C, /*reuse_a=*/false, /*reuse_b=*/false);

  if (lane == 0) {
    out[0] = D[0];
    out[1] = D[1];
    out[2] = D[2];
    out[3] = D[3];
    out[4] = D[4];
  }
}

extern "C" void kernel_launch(void* const* d_in, const int* in_sizes, int n_in,
                              void* d_out, int out_size, void* d_ws, size_t ws_size,
                              hipStream_t stream) {
  (void)in_sizes; (void)n_in; (void)out_size; (void)d_ws; (void)ws_size;
  const float* state  = (const float*)d_in[0];  // 5 floats
  const float* action = (const float*)d_in[1];  // 2 floats
  float* out = (float*)d_out;                   // 5 floats

  Dynamics_53884659696325_kernel<<<1, 32, 0, stream>>>(state, action, out);
}